// DecoderRNN_47665547051595
// MI455X (gfx1250) — compile-verified
//
#include <hip/hip_runtime.h>
#include <math.h>

#define Hdim   1024
#define Vdim   50000
#define MAXL   35

typedef __attribute__((ext_vector_type(2))) float v2f;
typedef __attribute__((ext_vector_type(8))) float v8f;

__device__ __forceinline__ float sigmoidf_(float v) { return 1.0f / (1.0f + expf(-v)); }

// -------------------------------------------------------------------------
// GEMV via V_WMMA_F32_16X16X4_F32 (full fp32 precision, 16 rows per wave).
// out[r] = act( dot(W[r,:], vec) + bias[r] )
// A-frag: lane m supplies (K0,K1), lane m+16 supplies (K2,K3) of the same row.
// We give lane m      W[row][k..k+3]   (global b128)
//         lane m+16   W[row][k+4..k+7]
// and B the matching vec chunks (LDS b128), broadcast across all 16 columns,
// so D columns are all identical & valid: c[q] = row (row0 + q + 8*half).
// -------------------------------------------------------------------------
__global__ __launch_bounds__(256) void gemv_wmma_f32(
    const float* __restrict__ W, const float* __restrict__ vec,
    const float* __restrict__ bias, float* __restrict__ out,
    int rows, int K, int relu)
{
    __shared__ __align__(16) float sv[2048];
    const int t = threadIdx.x;
    for (int i = t; i < K; i += 256) sv[i] = vec[i];
    __syncthreads();

    const int wave = t >> 5;
    const int lane = t & 31;
    const int half = lane >> 4;          // 0 -> (K0,K1), 1 -> (K2,K3)
    const int m    = lane & 15;          // row within 16-row tile
    const int tile = blockIdx.x * 8 + wave;
    const int row0 = tile * 16;
    if (row0 >= rows) return;            // wave-uniform: EXEC all-1 for WMMA

    const float* wrow = W + (size_t)(row0 + m) * (size_t)K + (size_t)(half * 4);
    const float* svh  = sv + half * 4;

    v8f c = {0.f, 0.f, 0.f, 0.f, 0.f, 0.f, 0.f, 0.f};
    for (int k = 0; k < K; k += 8) {
        const float4 a4 = *(const float4*)(wrow + k);   // global_load_b128
        const float4 b4 = *(const float4*)(svh + k);    // ds_load_b128
        v2f a, b;
        a.x = a4.x; a.y = a4.y; b.x = b4.x; b.y = b4.y;
        c = __builtin_amdgcn_wmma_f32_16x16x4_f32(false, a, false, b, (short)0, c, false, false);
        a.x = a4.z; a.y = a4.w; b.x = b4.z; b.y = b4.w;
        c = __builtin_amdgcn_wmma_f32_16x16x4_f32(false, a, false, b, (short)0, c, false, false);
    }

    // c[q] holds row (row0 + q + 8*half), replicated over the 16 lanes of each half.
    const int j = lane & 15;
    if (j < 8) {
        float val = 0.f;
        #pragma unroll
        for (int q = 0; q < 8; ++q) val = (j == q) ? c[q] : val;
        const int r = row0 + half * 8 + j;
        val += bias[r];
        if (relu) val = fmaxf(val, 0.f);
        out[r] = val;
    }
}

// -------------------------------------------------------------------------
// Attention front-end: embed gather, 35x2048 logits, softmax,
// attn_applied = attn_w @ encoder_outputs. One block of 256 (8 waves).
// -------------------------------------------------------------------------
__global__ __launch_bounds__(256) void attn_kernel(
    const int* __restrict__ x, const float* __restrict__ h,
    const float* __restrict__ enc, const float* __restrict__ emb,
    const float* __restrict__ W_attn, const float* __restrict__ b_attn,
    float* __restrict__ ws_emb,      // 1024  (first half of comb input)
    float* __restrict__ ws_attnapp,  // 1024  (second half of comb input)
    float* __restrict__ out_attnw)   // 35 -> d_out tail
{
    __shared__ __align__(16) float s_in[2 * Hdim];  // [emb_row | h]
    __shared__ float s_logit[MAXL];
    __shared__ float s_w[MAXL];

    const int t = threadIdx.x;
    const int xi = x[0];
    const float* erow = emb + (size_t)xi * Hdim;
    for (int i = t; i < Hdim; i += 256) {
        float e = erow[i];
        s_in[i] = e;
        ws_emb[i] = e;
        s_in[Hdim + i] = h[i];
    }
    __syncthreads();

    const int wave = t >> 5, lane = t & 31;
    for (int r = wave; r < MAXL; r += 8) {
        const float* wr = W_attn + (size_t)r * (2 * Hdim);
        float acc = 0.f;
        for (int k = lane; k < 2 * Hdim; k += 32) acc += wr[k] * s_in[k];
        #pragma unroll
        for (int off = 16; off > 0; off >>= 1) acc += __shfl_down(acc, off, 32);
        if (lane == 0) s_logit[r] = acc + b_attn[r];
    }
    __syncthreads();

    if (t == 0) {
        float M = -3.4e38f;
        for (int i = 0; i < MAXL; ++i) M = fmaxf(M, s_logit[i]);
        float S = 0.f;
        for (int i = 0; i < MAXL; ++i) { float e = expf(s_logit[i] - M); s_w[i] = e; S += e; }
        float inv = 1.0f / S;
        for (int i = 0; i < MAXL; ++i) s_w[i] *= inv;
    }
    __syncthreads();

    if (t < MAXL) out_attnw[t] = s_w[t];

    for (int jj = t; jj < Hdim; jj += 256) {
        float acc = 0.f;
        #pragma unroll 5
        for (int i = 0; i < MAXL; ++i) acc += s_w[i] * enc[(size_t)i * Hdim + jj];
        ws_attnapp[jj] = acc;
    }
}

// ------------------------- GRU gate elementwise ---------------------------
__global__ __launch_bounds__(256) void gru_gate_kernel(
    const float* __restrict__ xg, const float* __restrict__ hg,
    const float* __restrict__ h, float* __restrict__ hnew_ws,
    float* __restrict__ hnew_out)
{
    const int j = blockIdx.x * 256 + threadIdx.x;
    if (j >= Hdim) return;
    float r = sigmoidf_(xg[j] + hg[j]);
    float z = sigmoidf_(xg[Hdim + j] + hg[Hdim + j]);
    float n = tanhf(xg[2 * Hdim + j] + r * hg[2 * Hdim + j]);
    float hv = h[j];
    float o = (1.0f - z) * n + z * hv;
    hnew_ws[j] = o;
    hnew_out[j] = o;
}

// ------------------------- log-softmax tail -------------------------------
__global__ __launch_bounds__(256) void reduce_max_part(
    const float* __restrict__ v, int n, float* __restrict__ part)
{
    __shared__ float s[256];
    const int t = threadIdx.x;
    const int chunk = (n + gridDim.x - 1) / gridDim.x;
    const int lo = blockIdx.x * chunk;
    const int hi = min(lo + chunk, n);
    float m = -3.4e38f;
    for (int i = lo + t; i < hi; i += 256) m = fmaxf(m, v[i]);
    s[t] = m; __syncthreads();
    for (int o = 128; o > 0; o >>= 1) { if (t < o) s[t] = fmaxf(s[t], s[t + o]); __syncthreads(); }
    if (t == 0) part[blockIdx.x] = s[0];
}

__global__ __launch_bounds__(256) void finalize_max(
    const float* __restrict__ part, int np, float* __restrict__ scal)
{
    __shared__ float s[256];
    const int t = threadIdx.x;
    float m = -3.4e38f;
    for (int i = t; i < np; i += 256) m = fmaxf(m, part[i]);
    s[t] = m; __syncthreads();
    for (int o = 128; o > 0; o >>= 1) { if (t < o) s[t] = fmaxf(s[t], s[t + o]); __syncthreads(); }
    if (t == 0) scal[0] = s[0];
}

__global__ __launch_bounds__(256) void reduce_sumexp_part(
    const float* __restrict__ v, int n, const float* __restrict__ scal,
    float* __restrict__ part)
{
    __shared__ float s[256];
    const int t = threadIdx.x;
    const float M = scal[0];
    const int chunk = (n + gridDim.x - 1) / gridDim.x;
    const int lo = blockIdx.x * chunk;
    const int hi = min(lo + chunk, n);
    float a = 0.f;
    for (int i = lo + t; i < hi; i += 256) a += expf(v[i] - M);
    s[t] = a; __syncthreads();
    for (int o = 128; o > 0; o >>= 1) { if (t < o) s[t] += s[t + o]; __syncthreads(); }
    if (t == 0) part[blockIdx.x] = s[0];
}

__global__ __launch_bounds__(256) void finalize_sum(
    const float* __restrict__ part, int np, float* __restrict__ scal)
{
    __shared__ float s[256];
    const int t = threadIdx.x;
    float a = 0.f;
    for (int i = t; i < np; i += 256) a += part[i];
    s[t] = a; __syncthreads();
    for (int o = 128; o > 0; o >>= 1) { if (t < o) s[t] += s[t + o]; __syncthreads(); }
    if (t == 0) scal[1] = scal[0] + logf(s[0]);   // log-denominator
}

__global__ __launch_bounds__(256) void write_logp(
    const float* __restrict__ logits, const float* __restrict__ scal,
    float* __restrict__ out, int n)
{
    const int i = blockIdx.x * 256 + threadIdx.x;
    if (i < n) out[i] = logits[i] - scal[1];
}

// -------------------------------------------------------------------------
extern "C" void kernel_launch(void* const* d_in, const int* in_sizes, int n_in,
                              void* d_out, int out_size, void* d_ws, size_t ws_size,
                              hipStream_t stream)
{
    const int*   x       = (const int*)  d_in[0];
    const float* h_state = (const float*)d_in[1];   // [1,1,H]
    // d_in[2] (encoder_output) unused by the reference
    const float* enc     = (const float*)d_in[3];   // [35,H]
    const float* emb     = (const float*)d_in[4];   // [V,H]
    const float* W_attn  = (const float*)d_in[5];
    const float* b_attn  = (const float*)d_in[6];
    const float* W_comb  = (const float*)d_in[7];
    const float* b_comb  = (const float*)d_in[8];
    const float* W_ih    = (const float*)d_in[9];
    const float* b_ih    = (const float*)d_in[10];
    const float* W_hh    = (const float*)d_in[11];
    const float* b_hh    = (const float*)d_in[12];
    const float* W_out   = (const float*)d_in[13];
    const float* b_out   = (const float*)d_in[14];

    float* out = (float*)d_out;                 // [logp 50000 | h 1024 | attn_w 35]
    float* ws  = (float*)d_ws;

    // workspace layout (floats)
    float* ws_emb     = ws;                     // 1024 \ contiguous -> comb input (2048)
    float* ws_attnapp = ws + 1024;              // 1024 /
    float* ws_combvec = ws;                     // 2048
    float* ws_gruin   = ws + 2048;              // 1024
    float* ws_xg      = ws + 3072;              // 3072
    float* ws_hg      = ws + 6144;              // 3072
    float* ws_hnew    = ws + 9216;              // 1024
    float* ws_logits  = ws + 10240;             // 50000
    float* ws_red     = ws + 60240;             // 128
    float* ws_red2    = ws + 60368;             // 128
    float* ws_scal    = ws + 60496;             // 2

    float* out_logp  = out;
    float* out_h     = out + Vdim;
    float* out_attnw = out + Vdim + Hdim;

    // 1) embedding + attention weights + attn_applied
    attn_kernel<<<1, 256, 0, stream>>>(x, h_state, enc, emb, W_attn, b_attn,
                                       ws_emb, ws_attnapp, out_attnw);

    // 2) comb: [1024 x 2048] @ [emb|attn_applied] + b_comb, relu
    gemv_wmma_f32<<<(1024 / 16 + 7) / 8, 256, 0, stream>>>(
        W_comb, ws_combvec, b_comb, ws_gruin, 1024, 2048, 1);

    // 3) GRU gate pre-activations
    gemv_wmma_f32<<<(3072 / 16 + 7) / 8, 256, 0, stream>>>(
        W_ih, ws_gruin, b_ih, ws_xg, 3072, 1024, 0);
    gemv_wmma_f32<<<(3072 / 16 + 7) / 8, 256, 0, stream>>>(
        W_hh, h_state, b_hh, ws_hg, 3072, 1024, 0);

    // 4) GRU elementwise -> h_new
    gru_gate_kernel<<<(Hdim + 255) / 256, 256, 0, stream>>>(
        ws_xg, ws_hg, h_state, ws_hnew, out_h);

    // 5) big GEMV: logits = W_out @ h_new + b_out  (the 205 MB pass)
    gemv_wmma_f32<<<(Vdim / 16 + 7) / 8, 256, 0, stream>>>(
        W_out, ws_hnew, b_out, ws_logits, Vdim, 1024, 0);

    // 6) log_softmax tail
    reduce_max_part   <<<128, 256, 0, stream>>>(ws_logits, Vdim, ws_red);
    finalize_max      <<<1,   256, 0, stream>>>(ws_red, 128, ws_scal);
    reduce_sumexp_part<<<128, 256, 0, stream>>>(ws_logits, Vdim, ws_scal, ws_red2);
    finalize_sum      <<<1,   256, 0, stream>>>(ws_red2, 128, ws_scal);
    write_logp        <<<(Vdim + 255) / 256, 256, 0, stream>>>(ws_logits, ws_scal, out_logp, Vdim);
}